// DCNN_73787538145589
// MI455X (gfx1250) — compile-verified
//
#include <hip/hip_runtime.h>

#define BB 32
#define NN 1024
#define CIN 64
#define COUT 64
#define EE 16
#define KK 3
#define H1 16
#define M1 2

#define USE_ASYNC 1

#if defined(__has_builtin)
#if __has_builtin(__builtin_amdgcn_s_wait_asynccnt)
#define ASYNC_WAIT() __builtin_amdgcn_s_wait_asynccnt(0)
#endif
#endif
#ifndef ASYNC_WAIT
#define ASYNC_WAIT() asm volatile("s_wait_asynccnt 0x0" ::: "memory")
#endif

typedef float v2f __attribute__((ext_vector_type(2)));
typedef float v8f __attribute__((ext_vector_type(8)));

__device__ __forceinline__ float sigmoidf(float x) { return 1.0f / (1.0f + expf(-x)); }

// ---------------- Kernel 1: hypernetwork node vectors ----------------
__device__ __forceinline__ void mlp1(float s,
    const float* __restrict__ w1, const float* __restrict__ b1,
    const float* __restrict__ w2, const float* __restrict__ b2,
    const float* __restrict__ w3, const float* __restrict__ b3,
    float* outE)
{
    float h[H1];
#pragma unroll
    for (int j = 0; j < H1; ++j) h[j] = sigmoidf(s * w1[j] + b1[j]);
    float m[M1];
#pragma unroll
    for (int mm = 0; mm < M1; ++mm) {
        float a = b2[mm];
#pragma unroll
        for (int j = 0; j < H1; ++j) a += h[j] * w2[mm * H1 + j];
        m[mm] = sigmoidf(a);
    }
#pragma unroll
    for (int e = 0; e < EE; ++e) {
        float a = b3[e];
#pragma unroll
        for (int mm = 0; mm < M1; ++mm) a += m[mm] * w3[e * M1 + mm];
        outE[e] = a;
    }
}

__global__ void nv_kernel(
    const float* __restrict__ x, const float* __restrict__ emb0,
    const float* __restrict__ speed, const float* __restrict__ occupy,
    const float* __restrict__ fw1, const float* __restrict__ fb1,
    const float* __restrict__ fw2, const float* __restrict__ fb2,
    const float* __restrict__ fw3, const float* __restrict__ fb3,
    const float* __restrict__ gw1, const float* __restrict__ gb1,
    const float* __restrict__ gw2, const float* __restrict__ gb2,
    const float* __restrict__ gw3, const float* __restrict__ gb3,
    float* __restrict__ nv)
{
    int idx = blockIdx.x * blockDim.x + threadIdx.x;   // b*N + n
    if (idx >= BB * NN) return;
    const float* xr = x + (size_t)idx * CIN;
    float xl[CIN];
#pragma unroll
    for (int i = 0; i < CIN; ++i) xl[i] = xr[i];

    float h[H1];
#pragma unroll
    for (int j = 0; j < H1; ++j) {
        float a = fb1[j];
#pragma unroll
        for (int i = 0; i < CIN; ++i) a += xl[i] * fw1[j * CIN + i];
        h[j] = sigmoidf(a);
    }
    float m[M1];
#pragma unroll
    for (int mm = 0; mm < M1; ++mm) {
        float a = fb2[mm];
#pragma unroll
        for (int j = 0; j < H1; ++j) a += h[j] * fw2[mm * H1 + j];
        m[mm] = sigmoidf(a);
    }
    float filt[EE];
#pragma unroll
    for (int e = 0; e < EE; ++e) {
        float a = fb3[e];
#pragma unroll
        for (int mm = 0; mm < M1; ++mm) a += m[mm] * fw3[e * M1 + mm];
        filt[e] = a;
    }
    float f1[EE], f2[EE];
    mlp1(speed[idx],  gw1, gb1, gw2, gb2, gw3, gb3, f1);
    mlp1(occupy[idx], gw1, gb1, gw2, gb2, gw3, gb3, f2);
#pragma unroll
    for (int e = 0; e < EE; ++e) {
        float v = tanhf(emb0[(size_t)idx * EE + e] * filt[e]);
        v = tanhf(v * f1[e]);
        v = tanhf(v * f2[e]);
        nv[(size_t)idx * EE + e] = v;
    }
}

// ---------------- Kernel 2: fused supports-recompute diffusion --------
// Xout[b,m,c] = leaky_relu( sum_n relu(nv[b,m,:]·nv[b,n,:]) * Xin[b,n,c] )
#define XP 72   // X tile pitch (floats) — conflict-free halves
#define SP 17   // 16x16 tile pitch

__global__ void diffusion_kernel(const float* __restrict__ nv,
                                 const float* __restrict__ Xin,
                                 float* __restrict__ Xout)
{
    __shared__ __align__(16) float Xl[16 * XP];
    __shared__ float nvl[16 * SP];
    __shared__ float Sl[4][16 * SP];

    const int tid  = threadIdx.x;
    const int wave = tid >> 5;
    const int lane = tid & 31;
    const int col  = lane & 15;     // M for A-frags, N for B/D-frags
    const int half = lane >> 4;     // K-half selector
    const int b    = blockIdx.y;
    const int m0   = (blockIdx.x * 4 + wave) * 16;

    // preload this wave's nv rows (A operand of first GEMM)
    float anv[EE];
    {
        const float* p = nv + ((size_t)b * NN + m0 + col) * EE;
#pragma unroll
        for (int e = 0; e < EE; ++e) anv[e] = p[e];
    }

    v8f acc0 = {0,0,0,0,0,0,0,0};
    v8f acc1 = {0,0,0,0,0,0,0,0};
    v8f acc2 = {0,0,0,0,0,0,0,0};
    v8f acc3 = {0,0,0,0,0,0,0,0};

    for (int nt = 0; nt < NN / 16; ++nt) {
        const int n0 = nt * 16;
        __syncthreads();
#if USE_ASYNC
        // async global->LDS tile fill (bypasses VGPRs, tracked by ASYNCcnt)
        {
            const float* srcf = Xin + ((size_t)b * NN + n0) * CIN;
#pragma unroll
            for (int it = 0; it < 2; ++it) {
                int q = tid + it * 128;        // 0..255 float4s
                int r = q >> 4, c4 = q & 15;
                unsigned lo = (unsigned)(uintptr_t)&Xl[r * XP + c4 * 4];
                unsigned long long ga =
                    (unsigned long long)(uintptr_t)(srcf + r * CIN + c4 * 4);
                asm volatile("global_load_async_to_lds_b128 %0, %1, off"
                             :: "v"(lo), "v"(ga) : "memory");
            }
            const float* nsrc = nv + ((size_t)b * NN + n0) * EE;
#pragma unroll
            for (int it = 0; it < 2; ++it) {
                int q = tid + it * 128;
                int r = q >> 4, e = q & 15;
                unsigned lo = (unsigned)(uintptr_t)&nvl[r * SP + e];
                unsigned long long ga =
                    (unsigned long long)(uintptr_t)(nsrc + r * EE + e);
                asm volatile("global_load_async_to_lds_b32 %0, %1, off"
                             :: "v"(lo), "v"(ga) : "memory");
            }
            ASYNC_WAIT();   // drain this wave's ASYNCcnt before the barrier
        }
#else
        {
            const float4* src = (const float4*)(Xin + ((size_t)b * NN + n0) * CIN);
#pragma unroll
            for (int it = 0; it < 2; ++it) {
                int q = tid + it * 128;
                int r = q >> 4, c4 = q & 15;
                float4 v = src[r * 16 + c4];
                *(float4*)&Xl[r * XP + c4 * 4] = v;
            }
            const float* nsrc = nv + ((size_t)b * NN + n0) * EE;
#pragma unroll
            for (int it = 0; it < 2; ++it) {
                int q = tid + it * 128;
                int r = q >> 4, e = q & 15;
                nvl[r * SP + e] = nsrc[r * EE + e];
            }
        }
#endif
        __syncthreads();

        // S = relu(nv_m (16x16) @ nv_n^T (16x16)), K split into 4x4
        v8f s = {0,0,0,0,0,0,0,0};
#pragma unroll
        for (int kk = 0; kk < 4; ++kk) {
            v2f a, bf;
            a.x = half ? anv[4 * kk + 2] : anv[4 * kk + 0];
            a.y = half ? anv[4 * kk + 3] : anv[4 * kk + 1];
            const int e0 = 4 * kk + 2 * half;
            bf.x = nvl[col * SP + e0];       // B[e][n] = nv_n[n][e]
            bf.y = nvl[col * SP + e0 + 1];
            s = __builtin_amdgcn_wmma_f32_16x16x4_f32(false, a, false, bf,
                                                      (short)0, s, false, false);
        }
#pragma unroll
        for (int v = 0; v < 8; ++v) {
            float sv = s[v];
            Sl[wave][(v + 8 * half) * SP + col] = sv > 0.0f ? sv : 0.0f;
        }

        // Y += S (16x16) @ Xtile (16x64)
#pragma unroll
        for (int kk = 0; kk < 4; ++kk) {
            const int k0 = 4 * kk + 2 * half;
            v2f a;
            a.x = Sl[wave][col * SP + k0];
            a.y = Sl[wave][col * SP + k0 + 1];
            v2f b0, b1, b2, b3;
            b0.x = Xl[k0 * XP +  0 + col];  b0.y = Xl[(k0 + 1) * XP +  0 + col];
            b1.x = Xl[k0 * XP + 16 + col];  b1.y = Xl[(k0 + 1) * XP + 16 + col];
            b2.x = Xl[k0 * XP + 32 + col];  b2.y = Xl[(k0 + 1) * XP + 32 + col];
            b3.x = Xl[k0 * XP + 48 + col];  b3.y = Xl[(k0 + 1) * XP + 48 + col];
            acc0 = __builtin_amdgcn_wmma_f32_16x16x4_f32(false, a, false, b0, (short)0, acc0, false, false);
            acc1 = __builtin_amdgcn_wmma_f32_16x16x4_f32(false, a, false, b1, (short)0, acc1, false, false);
            acc2 = __builtin_amdgcn_wmma_f32_16x16x4_f32(false, a, false, b2, (short)0, acc2, false, false);
            acc3 = __builtin_amdgcn_wmma_f32_16x16x4_f32(false, a, false, b3, (short)0, acc3, false, false);
        }
    }

    // epilogue: leaky_relu (slope 0.01), store
    float* out = Xout + ((size_t)b * NN + m0) * COUT;
#pragma unroll
    for (int v = 0; v < 8; ++v) {
        const int row = v + 8 * half;
        float y0 = acc0[v]; y0 = y0 > 0.0f ? y0 : 0.01f * y0;
        float y1 = acc1[v]; y1 = y1 > 0.0f ? y1 : 0.01f * y1;
        float y2 = acc2[v]; y2 = y2 > 0.0f ? y2 : 0.01f * y2;
        float y3 = acc3[v]; y3 = y3 > 0.0f ? y3 : 0.01f * y3;
        out[row * COUT +  0 + col] = y0;
        out[row * COUT + 16 + col] = y1;
        out[row * COUT + 32 + col] = y2;
        out[row * COUT + 48 + col] = y3;
    }
}

// ---------------- Kernel 3: attention pooling + @weights1 -------------
__global__ void attn_kernel(const float* __restrict__ x0, const float* __restrict__ x1,
                            const float* __restrict__ x2,
                            const float* __restrict__ att_w, const float* __restrict__ att_b,
                            const float* __restrict__ w1, float* __restrict__ gf_out)
{
    __shared__ float sc[NN];
    __shared__ float red[256];
    __shared__ float gpart[4][COUT];
    __shared__ float gff[CIN];

    const int b = blockIdx.x, tid = threadIdx.x;
    const size_t base = (size_t)b * NN * CIN;
    const float third = 1.0f / 3.0f;

    float lmax = -1e30f;
    for (int n = tid; n < NN; n += 256) {
        const float* p0 = x0 + base + (size_t)n * CIN;
        const float* p1 = x1 + base + (size_t)n * CIN;
        const float* p2 = x2 + base + (size_t)n * CIN;
        float s = att_b[0];
#pragma unroll
        for (int c = 0; c < CIN; ++c) s += att_w[c] * (p0[c] + p1[c] + p2[c]) * third;
        sc[n] = s;
        lmax = fmaxf(lmax, s);
    }
    red[tid] = lmax;
    __syncthreads();
    for (int st = 128; st > 0; st >>= 1) {
        if (tid < st) red[tid] = fmaxf(red[tid], red[tid + st]);
        __syncthreads();
    }
    const float mx = red[0];
    __syncthreads();
    float lsum = 0.0f;
    for (int n = tid; n < NN; n += 256) lsum += expf(sc[n] - mx);
    red[tid] = lsum;
    __syncthreads();
    for (int st = 128; st > 0; st >>= 1) {
        if (tid < st) red[tid] += red[tid + st];
        __syncthreads();
    }
    const float Z = red[0];
    __syncthreads();

    const int g = tid >> 6, c = tid & 63;
    float part = 0.0f;
    for (int n = g; n < NN; n += 4) {
        float w = expf(sc[n] - mx);
        size_t o = base + (size_t)n * CIN + c;
        part += w * (x0[o] + x1[o] + x2[o]) * third;
    }
    gpart[g][c] = part;
    __syncthreads();
    if (tid < CIN) gff[c] = (gpart[0][c] + gpart[1][c] + gpart[2][c] + gpart[3][c]) / Z;
    __syncthreads();
    if (tid < COUT) {
        float v = 0.0f;
#pragma unroll
        for (int cc = 0; cc < CIN; ++cc) v += gff[cc] * w1[cc * COUT + tid];
        gf_out[b * COUT + tid] = v;
    }
}

// ---------------- Kernel 4: per-node pool weights + final combine -----
__global__ void final_kernel(
    const float* __restrict__ xA0, const float* __restrict__ xA1, const float* __restrict__ xA2,
    const float* __restrict__ xB0, const float* __restrict__ xB1, const float* __restrict__ xB2,
    const float* __restrict__ emb1,
    const float* __restrict__ wp, const float* __restrict__ bwp,
    const float* __restrict__ bias_pool,
    const float* __restrict__ alpha, const float* __restrict__ beta,
    const float* __restrict__ gf1, const float* __restrict__ gf2,
    float* __restrict__ out)
{
    __shared__ float wbuf[KK * CIN * COUT];   // 12288 floats = 48KB
    __shared__ float embs[EE];
    __shared__ float biasn[COUT];

    const int n = blockIdx.x, tid = threadIdx.x;
    if (tid < EE) embs[tid] = emb1[n * EE + tid];
    __syncthreads();
    if (tid < COUT) {
        float v = 0.0f;
#pragma unroll
        for (int e = 0; e < EE; ++e) v += embs[e] * bias_pool[e * COUT + tid];
        biasn[tid] = v;
    }
    // forward pool weights for this node: wbuf[(k*64+i)*64+o]
    for (int j = tid; j < KK * CIN * COUT; j += 256) {
        float v = 0.0f;
#pragma unroll
        for (int e = 0; e < EE; ++e) v += embs[e] * wp[e * (KK * CIN * COUT) + j];
        wbuf[j] = v;
    }
    __syncthreads();

    float acc1[8];
#pragma unroll
    for (int jj = 0; jj < 8; ++jj) {
        int idx = tid + 256 * jj;
        int b = idx >> 6, o = idx & 63;
        const float* p0 = xA0 + ((size_t)b * NN + n) * CIN;
        const float* p1 = xA1 + ((size_t)b * NN + n) * CIN;
        const float* p2 = xA2 + ((size_t)b * NN + n) * CIN;
        float a = 0.0f;
#pragma unroll
        for (int i = 0; i < CIN; ++i) {
            a += p0[i] * wbuf[(0 * CIN + i) * COUT + o];
            a += p1[i] * wbuf[(1 * CIN + i) * COUT + o];
            a += p2[i] * wbuf[(2 * CIN + i) * COUT + o];
        }
        acc1[jj] = a;
    }
    __syncthreads();
    // backward pool weights
    for (int j = tid; j < KK * CIN * COUT; j += 256) {
        float v = 0.0f;
#pragma unroll
        for (int e = 0; e < EE; ++e) v += embs[e] * bwp[e * (KK * CIN * COUT) + j];
        wbuf[j] = v;
    }
    __syncthreads();

    float acc2[8];
#pragma unroll
    for (int jj = 0; jj < 8; ++jj) {
        int idx = tid + 256 * jj;
        int b = idx >> 6, o = idx & 63;
        const float* p0 = xB0 + ((size_t)b * NN + n) * CIN;
        const float* p1 = xB1 + ((size_t)b * NN + n) * CIN;
        const float* p2 = xB2 + ((size_t)b * NN + n) * CIN;
        float a = 0.0f;
#pragma unroll
        for (int i = 0; i < CIN; ++i) {
            a += p0[i] * wbuf[(0 * CIN + i) * COUT + o];
            a += p1[i] * wbuf[(1 * CIN + i) * COUT + o];
            a += p2[i] * wbuf[(2 * CIN + i) * COUT + o];
        }
        acc2[jj] = a;
    }

    const float al = alpha[0], be = beta[0];
#pragma unroll
    for (int jj = 0; jj < 8; ++jj) {
        int idx = tid + 256 * jj;
        int b = idx >> 6, o = idx & 63;
        out[((size_t)b * NN + n) * COUT + o] =
            al * acc1[jj] + be * acc2[jj] + biasn[o] + gf1[b * COUT + o] + gf2[b * COUT + o];
    }
}

// ---------------------------------------------------------------------
extern "C" void kernel_launch(void* const* d_in, const int* in_sizes, int n_in,
                              void* d_out, int out_size, void* d_ws, size_t ws_size,
                              hipStream_t stream)
{
    (void)in_sizes; (void)n_in; (void)out_size; (void)ws_size;
    const float* x      = (const float*)d_in[0];
    const float* emb0   = (const float*)d_in[1];
    const float* emb1   = (const float*)d_in[2];
    const float* speed  = (const float*)d_in[5];
    const float* occupy = (const float*)d_in[6];
    const float* fw1 = (const float*)d_in[7];
    const float* fb1 = (const float*)d_in[8];
    const float* fw2 = (const float*)d_in[9];
    const float* fb2 = (const float*)d_in[10];
    const float* fw3 = (const float*)d_in[11];
    const float* fb3 = (const float*)d_in[12];
    const float* gw1 = (const float*)d_in[13];
    const float* gb1 = (const float*)d_in[14];
    const float* gw2 = (const float*)d_in[15];
    const float* gb2 = (const float*)d_in[16];
    const float* gw3 = (const float*)d_in[17];
    const float* gb3 = (const float*)d_in[18];
    const float* alpha = (const float*)d_in[19];
    const float* beta  = (const float*)d_in[20];
    const float* wp    = (const float*)d_in[21];
    const float* bwp   = (const float*)d_in[22];
    const float* bias_pool = (const float*)d_in[23];
    const float* w1    = (const float*)d_in[24];
    const float* att_w = (const float*)d_in[25];
    const float* att_b = (const float*)d_in[26];
    float* out = (float*)d_out;

    float* ws  = (float*)d_ws;
    float* nv  = ws;                         // B*N*E   = 524288
    float* xs1 = nv  + (size_t)BB * NN * EE;
    float* xs2 = xs1 + (size_t)BB * NN * CIN;
    float* xb1 = xs2 + (size_t)BB * NN * CIN;
    float* xb2 = xb1 + (size_t)BB * NN * CIN;
    float* gf1 = xb2 + (size_t)BB * NN * CIN;
    float* gf2 = gf1 + (size_t)BB * COUT;

    nv_kernel<<<(BB * NN) / 256, 256, 0, stream>>>(
        x, emb0, speed, occupy,
        fw1, fb1, fw2, fb2, fw3, fb3,
        gw1, gb1, gw2, gb2, gw3, gb3, nv);

    dim3 dgrid(NN / 64, BB);
    diffusion_kernel<<<dgrid, 128, 0, stream>>>(nv, x,   xs1);
    diffusion_kernel<<<dgrid, 128, 0, stream>>>(nv, xs1, xs2);
    diffusion_kernel<<<dgrid, 128, 0, stream>>>(nv, xs2, xb1);
    diffusion_kernel<<<dgrid, 128, 0, stream>>>(nv, xb1, xb2);

    attn_kernel<<<BB, 256, 0, stream>>>(x,   xs1, xs2, att_w, att_b, w1, gf1);
    attn_kernel<<<BB, 256, 0, stream>>>(xs2, xb1, xb2, att_w, att_b, w1, gf2);

    final_kernel<<<NN, 256, 0, stream>>>(
        x, xs1, xs2, xs2, xb1, xb2,
        emb1, wp, bwp, bias_pool, alpha, beta, gf1, gf2, out);
}